// RankChannels_38362647888217
// MI455X (gfx1250) — compile-verified
//
#include <hip/hip_runtime.h>
#include <hip/hip_bf16.h>
#include <math.h>

typedef __attribute__((ext_vector_type(2))) float v2f;
typedef __attribute__((ext_vector_type(4))) float v4f;
typedef __attribute__((ext_vector_type(8))) float v8f;

#define NCH   512
#define HW    65536          // 256*256 elements per channel
#define TOPK  10

// One block per channel: stream 65536 floats with b128 loads, reduce via
// V_WMMA_F32_16X16X4_F32 against an all-ones B (D[m][n] = sum_k A[m][k]).
// Every loaded float lands in exactly one A slot, so sum(D)/16 == sum(data).
__global__ void __launch_bounds__(256)
chanmean_wmma_kernel(const float* __restrict__ x, float* __restrict__ means) {
    const int c = blockIdx.x;          // channel 0..511
    const int t = threadIdx.x;         // 0..255
    const v4f* __restrict__ p = (const v4f*)(x + (size_t)c * HW);

    v8f acc0 = {};                     // two independent accumulator chains
    v8f acc1 = {};
    v2f ones; ones.x = 1.0f; ones.y = 1.0f;

    // 65536 floats / block = 16384 float4; 256 threads -> 64 float4 each,
    // fully coalesced: iteration i, thread t reads float4 index i*256+t.
    #pragma unroll 8
    for (int i = 0; i < 64; ++i) {
        v4f d = p[i * 256 + t];
        v2f a0; a0.x = d.x; a0.y = d.y;
        v2f a1; a1.x = d.z; a1.y = d.w;
        // 8 args: (neg_a, A, neg_b, B, c_mod, C, reuse_a, reuse_b)
        acc0 = __builtin_amdgcn_wmma_f32_16x16x4_f32(
            false, a0, false, ones, (short)0, acc0, false, false);
        acc1 = __builtin_amdgcn_wmma_f32_16x16x4_f32(
            false, a1, false, ones, (short)0, acc1, false, false);
    }

    // Per-lane partial: 8 D components of each accumulator.
    float s = 0.0f;
    #pragma unroll
    for (int j = 0; j < 8; ++j) s += acc0[j] + acc1[j];

    // Wave32 butterfly reduce.
    #pragma unroll
    for (int off = 16; off > 0; off >>= 1)
        s += __shfl_xor(s, off, 32);

    __shared__ float wsum[8];
    const int wave = t >> 5, lane = t & 31;
    if (lane == 0) wsum[wave] = s;
    __syncthreads();
    if (t == 0) {
        float tot = 0.0f;
        #pragma unroll
        for (int w = 0; w < 8; ++w) tot += wsum[w];
        // /16: ones-B replicates each row-sum across 16 columns; /65536: mean.
        means[c] = tot * (1.0f / (16.0f * (float)HW));
    }
}

// Single wave32: iterative argmax over 512 means, tie-break toward the lower
// index (matches jax.lax.top_k stability), evict winner with -inf.
__global__ void __launch_bounds__(32)
topk_kernel(const float* __restrict__ means, int* __restrict__ out) {
    __shared__ float sval[NCH];
    const int lane = threadIdx.x;      // 0..31

    for (int i = lane; i < NCH; i += 32) sval[i] = means[i];
    __syncthreads();

    for (int k = 0; k < TOPK; ++k) {
        float bv = -INFINITY;
        int   bi = 0x7fffffff;
        // Each lane owns channels [lane*16, lane*16+16).
        #pragma unroll
        for (int j = 0; j < 16; ++j) {
            const int idx = lane * 16 + j;
            const float v = sval[idx];
            if (v > bv || (v == bv && idx < bi)) { bv = v; bi = idx; }
        }
        // Wave butterfly with (max value, min index on tie) comparator.
        #pragma unroll
        for (int off = 16; off > 0; off >>= 1) {
            const float ov = __shfl_xor(bv, off, 32);
            const int   oi = __shfl_xor(bi, off, 32);
            if (ov > bv || (ov == bv && oi < bi)) { bv = ov; bi = oi; }
        }
        if (lane == 0) out[k] = bi;
        if (lane == (bi >> 4)) sval[bi] = -INFINITY;  // owner lane evicts
        __syncthreads();
    }
}

extern "C" void kernel_launch(void* const* d_in, const int* in_sizes, int n_in,
                              void* d_out, int out_size, void* d_ws, size_t ws_size,
                              hipStream_t stream) {
    const float* x   = (const float*)d_in[0];   // [1,512,256,256] fp32
    float* means     = (float*)d_ws;            // 512 floats of scratch
    int*   out       = (int*)d_out;             // 10 int32 indices

    chanmean_wmma_kernel<<<NCH, 256, 0, stream>>>(x, means);
    topk_kernel<<<1, 32, 0, stream>>>(means, out);
}